// RoiPoolingConv_23433341567158
// MI455X (gfx1250) — compile-verified
//
#include <hip/hip_runtime.h>
#include <hip/hip_bf16.h>

#define P 7
#define NUM_ROIS 300
#define IMG_H 1024
#define IMG_W 1024
#define IMG_C 256

#if defined(__gfx1250__) &&                                           \
    __has_builtin(__builtin_amdgcn_global_load_async_to_lds_b128) &&  \
    __has_builtin(__builtin_amdgcn_s_wait_asynccnt)
#define USE_ASYNC 1
#define GLOBAL_AS __attribute__((address_space(1)))
#define LDS_AS __attribute__((address_space(3)))
typedef int v4i __attribute__((ext_vector_type(4)));
typedef GLOBAL_AS v4i* gv4i_p;
typedef LDS_AS v4i* lv4i_p;
#else
#define USE_ASYNC 0
#endif

// One block = one (roi, py) output row. 64 threads (2 waves), each lane owns 4
// channels. All 7 cells' corner lines (28 async b128 transfers = 14 KB/wave)
// are issued up front through the CDNA5 async global->LDS engine, then drained
// in order with descending s_wait_asynccnt — max memory-level parallelism for
// a pure-bandwidth kernel (~74 MB total traffic, ~3 us at 23.3 TB/s).
__global__ __launch_bounds__(64) void roi_pool_kernel(
    const float* __restrict__ img,   // (1, H, W, C)
    const float* __restrict__ rois,  // (1, R, 4) float (x0, y0, w, h)
    float* __restrict__ out)         // (1, R, P, P, C)
{
#if USE_ASYNC
    __shared__ float stage[P][4][IMG_C];  // 28 KB: 7 cells x 4 corners x 256 ch
#endif
    const int blk = blockIdx.x;
    const int r   = blk / P;
    const int py  = blk % P;
    const int tid = threadIdx.x;
    const int ch  = tid << 2;  // 4 channels per lane

    // ROI params (uniform across block -> compiler scalarizes)
    const float rx = rois[r * 4 + 0];
    const float ry = rois[r * 4 + 1];
    const float rw = rois[r * 4 + 2];
    const float rh = rois[r * 4 + 3];
    const int x0 = (int)rx, y0 = (int)ry, w = (int)rw, h = (int)rh;
    const float wf = (float)w, hf = (float)h;

    // y coordinates for this pooled row (match JAX: (i+0.5)*(size/P)-0.5, clipped)
    float cy = ((float)py + 0.5f) * (hf / 7.0f) - 0.5f;
    cy = fminf(fmaxf(cy, 0.0f), hf - 1.0f);
    const int   ylo = (int)floorf(cy);
    const int   yhi = min(ylo + 1, h - 1);
    const float fy  = cy - (float)ylo;
    const float gy  = 1.0f - fy;

    const size_t rowLo = (size_t)(y0 + ylo) * IMG_W * IMG_C;
    const size_t rowHi = (size_t)(y0 + yhi) * IMG_W * IMG_C;

    auto xcoords = [&](int px, int& XL, int& XH, float& fx) {
        float cx = ((float)px + 0.5f) * (wf / 7.0f) - 0.5f;
        cx = fminf(fmaxf(cx, 0.0f), wf - 1.0f);
        int xlo = (int)floorf(cx);
        int xhi = min(xlo + 1, w - 1);
        fx = cx - (float)xlo;
        XL = x0 + xlo;
        XH = x0 + xhi;
    };

#if USE_ASYNC
    float fxs[P];

    // Issue phase: 28 in-flight async transfers (ASYNCcnt max is 63).
    #pragma unroll
    for (int px = 0; px < P; ++px) {
        int XL, XH;
        xcoords(px, XL, XH, fxs[px]);
        const float* g0 = img + rowLo + (size_t)XL * IMG_C + ch;
        const float* g1 = img + rowLo + (size_t)XH * IMG_C + ch;
        const float* g2 = img + rowHi + (size_t)XL * IMG_C + ch;
        const float* g3 = img + rowHi + (size_t)XH * IMG_C + ch;
        __builtin_amdgcn_global_load_async_to_lds_b128(
            (gv4i_p)(v4i*)g0, (lv4i_p)(v4i*)&stage[px][0][ch], 0, 0);
        __builtin_amdgcn_global_load_async_to_lds_b128(
            (gv4i_p)(v4i*)g1, (lv4i_p)(v4i*)&stage[px][1][ch], 0, 0);
        __builtin_amdgcn_global_load_async_to_lds_b128(
            (gv4i_p)(v4i*)g2, (lv4i_p)(v4i*)&stage[px][2][ch], 0, 0);
        __builtin_amdgcn_global_load_async_to_lds_b128(
            (gv4i_p)(v4i*)g3, (lv4i_p)(v4i*)&stage[px][3][ch], 0, 0);
    }

    // Drain phase: async loads retire in order, so waiting for <= 24-4*px
    // outstanding guarantees cell px's 4 corner lines are in LDS.
    auto cell = [&](int px) {
        const float fx = fxs[px];
        const float gx = 1.0f - fx;
        const float4 tl = *(const float4*)&stage[px][0][ch];
        const float4 tr = *(const float4*)&stage[px][1][ch];
        const float4 bl = *(const float4*)&stage[px][2][ch];
        const float4 br = *(const float4*)&stage[px][3][ch];
        float4 o;
        o.x = (tl.x * gx + tr.x * fx) * gy + (bl.x * gx + br.x * fx) * fy;
        o.y = (tl.y * gx + tr.y * fx) * gy + (bl.y * gx + br.y * fx) * fy;
        o.z = (tl.z * gx + tr.z * fx) * gy + (bl.z * gx + br.z * fx) * fy;
        o.w = (tl.w * gx + tr.w * fx) * gy + (bl.w * gx + br.w * fx) * fy;
        const size_t off = ((((size_t)r * P + py) * P + px) * IMG_C) + ch;
        *(float4*)(out + off) = o;
    };

    __builtin_amdgcn_s_wait_asynccnt(24); asm volatile("" ::: "memory"); cell(0);
    __builtin_amdgcn_s_wait_asynccnt(20); asm volatile("" ::: "memory"); cell(1);
    __builtin_amdgcn_s_wait_asynccnt(16); asm volatile("" ::: "memory"); cell(2);
    __builtin_amdgcn_s_wait_asynccnt(12); asm volatile("" ::: "memory"); cell(3);
    __builtin_amdgcn_s_wait_asynccnt(8);  asm volatile("" ::: "memory"); cell(4);
    __builtin_amdgcn_s_wait_asynccnt(4);  asm volatile("" ::: "memory"); cell(5);
    __builtin_amdgcn_s_wait_asynccnt(0);  asm volatile("" ::: "memory"); cell(6);
#else
    // Fallback: direct coalesced b128 loads (no async builtins on this toolchain)
    for (int px = 0; px < P; ++px) {
        int XL, XH;
        float fx;
        xcoords(px, XL, XH, fx);
        const float4 tl = *(const float4*)(img + rowLo + (size_t)XL * IMG_C + ch);
        const float4 tr = *(const float4*)(img + rowLo + (size_t)XH * IMG_C + ch);
        const float4 bl = *(const float4*)(img + rowHi + (size_t)XL * IMG_C + ch);
        const float4 br = *(const float4*)(img + rowHi + (size_t)XH * IMG_C + ch);

        const float gx = 1.0f - fx;
        float4 o;
        o.x = (tl.x * gx + tr.x * fx) * gy + (bl.x * gx + br.x * fx) * fy;
        o.y = (tl.y * gx + tr.y * fx) * gy + (bl.y * gx + br.y * fx) * fy;
        o.z = (tl.z * gx + tr.z * fx) * gy + (bl.z * gx + br.z * fx) * fy;
        o.w = (tl.w * gx + tr.w * fx) * gy + (bl.w * gx + br.w * fx) * fy;

        const size_t off = ((((size_t)r * P + py) * P + px) * IMG_C) + ch;
        *(float4*)(out + off) = o;
    }
#endif
}

extern "C" void kernel_launch(void* const* d_in, const int* in_sizes, int n_in,
                              void* d_out, int out_size, void* d_ws, size_t ws_size,
                              hipStream_t stream) {
    (void)in_sizes; (void)n_in; (void)out_size; (void)d_ws; (void)ws_size;
    const float* img  = (const float*)d_in[0];
    const float* rois = (const float*)d_in[1];
    float* out = (float*)d_out;
    // 300 ROIs * 7 pooled rows = 2100 blocks of 64 threads (2 waves each)
    roi_pool_kernel<<<NUM_ROIS * P, 64, 0, stream>>>(img, rois, out);
}